// SelectiveStateSpaceModel_3607772529041
// MI455X (gfx1250) — compile-verified
//
#include <hip/hip_runtime.h>

// Selective SSM (Mamba-like) for MI455X / gfx1250, wave32.
// B=2, L=2048, D=1024, N=16, R=64, DBC=R+2N=96.
//
// Pipeline:
//   k_dbc   : dbc = x @ W_dbc^T + b_dbc                     (f32 WMMA 16x16x4)
//   k_delta : delta = softplus(dbc[:, :64] @ W_up^T + b_up) (f32 WMMA 16x16x4)
//   k_pass1 : per-chunk scan (h0=0) -> aprod[16], hend[16] per (b,d,chunk)
//   k_pass2 : chain chunk boundaries -> hinit per (b,d,chunk)
//   k_pass3 : re-scan each chunk from hinit, emit y = sum_n h*Cm (register dot)
//
// CDNA5 specifics: wave32 WMMA f32 16x16x4 for the GEMMs; per-block Bm/Cm
// slabs staged to LDS with global_load_async_to_lds_b128 (ASYNCcnt) so the
// scan inner loop's vmem is only the two coalesced HBM streams (delta, x).

typedef __attribute__((ext_vector_type(2))) float v2f;
typedef __attribute__((ext_vector_type(8))) float v8f;

#define BL_   4096   // B*L
#define D_    1024
#define N_    16
#define R_    64
#define DBC_  96
#define L_    2048
#define B_    2
#define CH_   32     // chunks
#define CL_   64     // chunk length (L_/CH_)

// Async global->LDS copy of one 16-byte segment (gfx1250, ASYNCcnt-tracked).
__device__ __forceinline__ void async_b128_to_lds(unsigned lds_byte_off,
                                                  const void* gaddr) {
  asm volatile("global_load_async_to_lds_b128 %0, %1, off"
               :: "v"(lds_byte_off), "v"(gaddr) : "memory");
}
__device__ __forceinline__ void wait_asynccnt0() {
  asm volatile("s_wait_asynccnt 0x0" ::: "memory");
}

// ---------------------------------------------------------------------------
// Kernel A: dbc[4096,96] = x[4096,1024] @ W_dbc[96,1024]^T + b_dbc
// One wave computes a 16(M) x 96(N) strip; K in steps of 4 (f32 WMMA).
// A-frag (16x4 f32): lane m=lane&15, VGPR0/1 = K(koff)/K(koff+1),
// koff=(lane>=16)*2.  B-frag mirrors with lane = N.  C/D: M = v+8*(lane>=16).
// ---------------------------------------------------------------------------
__global__ void k_dbc(const float* __restrict__ x,
                      const float* __restrict__ Wdbc,
                      const float* __restrict__ bdbc,
                      float* __restrict__ dbc) {
  const int lane  = threadIdx.x & 31;
  const int wave  = threadIdx.x >> 5;
  const int mtile = blockIdx.x * 4 + wave;          // 0..255
  const int m     = lane & 15;
  const int koff  = (lane >> 4) * 2;

  const float* __restrict__ xrow = x + (size_t)(mtile * 16 + m) * D_;

  v8f acc[6] = {};

#pragma unroll 4
  for (int k = 0; k < D_; k += 4) {
    v2f a;
    a.x = xrow[k + koff];
    a.y = xrow[k + koff + 1];
#pragma unroll
    for (int j = 0; j < 6; ++j) {
      const int n = j * 16 + m;
      v2f bfrag;
      bfrag.x = Wdbc[(size_t)n * D_ + k + koff];
      bfrag.y = Wdbc[(size_t)n * D_ + k + koff + 1];
      acc[j] = __builtin_amdgcn_wmma_f32_16x16x4_f32(
          false, a, false, bfrag, (short)0, acc[j], false, false);
    }
  }

  const int row0  = mtile * 16;
  const int rhalf = (lane >> 4) * 8;
#pragma unroll
  for (int j = 0; j < 6; ++j) {
    const int col = j * 16 + m;
    const float bias = bdbc[col];
#pragma unroll
    for (int v = 0; v < 8; ++v) {
      dbc[(size_t)(row0 + rhalf + v) * DBC_ + col] = acc[j][v] + bias;
    }
  }
}

// ---------------------------------------------------------------------------
// Kernel B: delta[4096,1024] = softplus(dbc[:, :64] @ W_up[1024,64]^T + b_up)
// ---------------------------------------------------------------------------
__global__ void k_delta(const float* __restrict__ dbc,
                        const float* __restrict__ Wup,
                        const float* __restrict__ bup,
                        float* __restrict__ delta) {
  const int lane  = threadIdx.x & 31;
  const int wave  = threadIdx.x >> 5;
  const int mtile = blockIdx.x * 4 + wave;           // 0..255
  const int ncol0 = blockIdx.y * 64;
  const int m     = lane & 15;
  const int koff  = (lane >> 4) * 2;

  const float* __restrict__ arow = dbc + (size_t)(mtile * 16 + m) * DBC_;

  v8f acc[4] = {};

#pragma unroll
  for (int k = 0; k < R_; k += 4) {
    v2f a;
    a.x = arow[k + koff];
    a.y = arow[k + koff + 1];
#pragma unroll
    for (int j = 0; j < 4; ++j) {
      const int n = ncol0 + j * 16 + m;
      v2f bfrag;
      bfrag.x = Wup[(size_t)n * R_ + k + koff];
      bfrag.y = Wup[(size_t)n * R_ + k + koff + 1];
      acc[j] = __builtin_amdgcn_wmma_f32_16x16x4_f32(
          false, a, false, bfrag, (short)0, acc[j], false, false);
    }
  }

  const int row0  = mtile * 16;
  const int rhalf = (lane >> 4) * 8;
#pragma unroll
  for (int j = 0; j < 4; ++j) {
    const int col = ncol0 + j * 16 + m;
    const float bias = bup[col];
#pragma unroll
    for (int v = 0; v < 8; ++v) {
      float z = acc[j][v] + bias;
      float sp = (z > 20.f) ? z : log1pf(__expf(z));   // stable softplus
      delta[(size_t)(row0 + rhalf + v) * D_ + col] = sp;
    }
  }
}

// ---------------------------------------------------------------------------
// Scan pass 1: thread = (b, d, chunk), g = ((b*CH + c)*D + d).
// Block of 256 threads shares one (b,c): stage the chunk's Bm rows
// (CL x 16 f32 = 4KB) to LDS via async loads, then scan with h0=0.
// ---------------------------------------------------------------------------
__global__ void k_pass1(const float* __restrict__ x,
                        const float* __restrict__ A_log,
                        const float* __restrict__ dbc,
                        const float* __restrict__ delta,
                        float* __restrict__ aprod_ws,
                        float* __restrict__ hend_ws) {
  __shared__ float sBm[CL_ * N_];                       // 4KB

  const int g = blockIdx.x * blockDim.x + threadIdx.x;  // 0..65535
  const int d = g & (D_ - 1);
  const int rest = g >> 10;
  const int c = rest & (CH_ - 1);
  const int b = rest >> 5;
  const int l0 = c * CL_;
  const size_t r0 = (size_t)b * L_ + l0;

  // ---- async stage: 256 b128 segments, one per thread ----
  {
    const int s = threadIdx.x;                          // 0..255
    const int lr = s >> 2;                              // row within chunk
    const int part = s & 3;                             // 4 x b128 per 16-float row
    const void* gsrc = (const void*)(dbc + (r0 + lr) * DBC_ + R_ + part * 4);
    const unsigned loff = (unsigned)(size_t)&sBm[s * 4];
    async_b128_to_lds(loff, gsrc);
    wait_asynccnt0();
  }
  __syncthreads();

  float A[N_], h[N_], ap[N_];
  {
    const float4* al = (const float4*)(A_log + (size_t)d * N_);
#pragma unroll
    for (int q = 0; q < 4; ++q) {
      float4 v = al[q];
      A[4*q+0] = -__expf(v.x); A[4*q+1] = -__expf(v.y);
      A[4*q+2] = -__expf(v.z); A[4*q+3] = -__expf(v.w);
    }
  }
#pragma unroll
  for (int n = 0; n < N_; ++n) { h[n] = 0.f; ap[n] = 1.f; }

  for (int t = 0; t < CL_; ++t) {
    const size_t r = r0 + t;
    const float dl = delta[r * D_ + d];
    const float xv = x[r * D_ + d];
    const float4* bm = (const float4*)&sBm[t * N_];     // ds_load_b128
    float Bm[N_];
#pragma unroll
    for (int q = 0; q < 4; ++q) {
      float4 v = bm[q];
      Bm[4*q+0] = v.x; Bm[4*q+1] = v.y; Bm[4*q+2] = v.z; Bm[4*q+3] = v.w;
    }
    const float dlx = dl * xv;
#pragma unroll
    for (int n = 0; n < N_; ++n) {
      const float a = __expf(dl * A[n]);
      ap[n] *= a;
      h[n] = fmaf(a, h[n], Bm[n] * dlx);
    }
  }

  const size_t base = ((size_t)(b * D_ + d) * CH_ + c) * N_;
  float4* apo = (float4*)(aprod_ws + base);
  float4* heo = (float4*)(hend_ws + base);
#pragma unroll
  for (int q = 0; q < 4; ++q) {
    apo[q] = make_float4(ap[4*q+0], ap[4*q+1], ap[4*q+2], ap[4*q+3]);
    heo[q] = make_float4(h[4*q+0],  h[4*q+1],  h[4*q+2],  h[4*q+3]);
  }
}

// ---------------------------------------------------------------------------
// Scan pass 2: thread = (b, d, n). Chain CH_ chunk summaries:
//   hinit[c] = h;  h = aprod[c]*h + hend[c]
// ---------------------------------------------------------------------------
__global__ void k_pass2(const float* __restrict__ aprod_ws,
                        const float* __restrict__ hend_ws,
                        float* __restrict__ hinit_ws) {
  const int g = blockIdx.x * blockDim.x + threadIdx.x;  // 0..32767
  const int n = g & (N_ - 1);
  const int p = g >> 4;                                 // b*D + d

  float h = 0.f;
  const size_t pb = (size_t)p * CH_ * N_;
  for (int c = 0; c < CH_; ++c) {
    const size_t idx = pb + (size_t)c * N_ + n;
    hinit_ws[idx] = h;
    h = fmaf(aprod_ws[idx], h, hend_ws[idx]);
  }
}

// ---------------------------------------------------------------------------
// Scan pass 3: layout as pass 1; stage Bm+Cm (CL x 32 f32 = 8KB) to LDS,
// seed h from hinit, emit y[b,l,d] = sum_n h[n]*Cm[n] (in-register dot).
// ---------------------------------------------------------------------------
__global__ void k_pass3(const float* __restrict__ x,
                        const float* __restrict__ A_log,
                        const float* __restrict__ dbc,
                        const float* __restrict__ delta,
                        const float* __restrict__ hinit_ws,
                        float* __restrict__ y) {
  __shared__ float sBC[CL_ * 2 * N_];                   // 8KB: per row Bm[16]|Cm[16]

  const int g = blockIdx.x * blockDim.x + threadIdx.x;  // 0..65535
  const int d = g & (D_ - 1);
  const int rest = g >> 10;
  const int c = rest & (CH_ - 1);
  const int b = rest >> 5;
  const int l0 = c * CL_;
  const size_t r0 = (size_t)b * L_ + l0;

  // ---- async stage: 512 b128 segments, two per thread ----
  {
#pragma unroll
    for (int i = 0; i < 2; ++i) {
      const int s = threadIdx.x + i * 256;              // 0..511
      const int lr = s >> 3;                            // row within chunk
      const int part = s & 7;                           // 8 x b128 per 32-float row
      const void* gsrc = (const void*)(dbc + (r0 + lr) * DBC_ + R_ + part * 4);
      const unsigned loff = (unsigned)(size_t)&sBC[s * 4];
      async_b128_to_lds(loff, gsrc);
    }
    wait_asynccnt0();
  }
  __syncthreads();

  float A[N_], h[N_];
  {
    const float4* al = (const float4*)(A_log + (size_t)d * N_);
#pragma unroll
    for (int q = 0; q < 4; ++q) {
      float4 v = al[q];
      A[4*q+0] = -__expf(v.x); A[4*q+1] = -__expf(v.y);
      A[4*q+2] = -__expf(v.z); A[4*q+3] = -__expf(v.w);
    }
  }
  {
    const size_t base = ((size_t)(b * D_ + d) * CH_ + c) * N_;
    const float4* hi = (const float4*)(hinit_ws + base);
#pragma unroll
    for (int q = 0; q < 4; ++q) {
      float4 v = hi[q];
      h[4*q+0] = v.x; h[4*q+1] = v.y; h[4*q+2] = v.z; h[4*q+3] = v.w;
    }
  }

  for (int t = 0; t < CL_; ++t) {
    const size_t r = r0 + t;
    const float dl = delta[r * D_ + d];
    const float xv = x[r * D_ + d];
    const float4* bc = (const float4*)&sBC[t * 2 * N_]; // ds_load_b128 x8
    float Bm[N_], Cm[N_];
#pragma unroll
    for (int q = 0; q < 4; ++q) {
      float4 v = bc[q];
      Bm[4*q+0] = v.x; Bm[4*q+1] = v.y; Bm[4*q+2] = v.z; Bm[4*q+3] = v.w;
      float4 w = bc[q + 4];
      Cm[4*q+0] = w.x; Cm[4*q+1] = w.y; Cm[4*q+2] = w.z; Cm[4*q+3] = w.w;
    }
    const float dlx = dl * xv;
    float acc = 0.f;
#pragma unroll
    for (int n = 0; n < N_; ++n) {
      const float a = __expf(dl * A[n]);
      h[n] = fmaf(a, h[n], Bm[n] * dlx);
      acc = fmaf(h[n], Cm[n], acc);
    }
    y[r * D_ + d] = acc;                                // coalesced across lanes
  }
}

// ---------------------------------------------------------------------------
extern "C" void kernel_launch(void* const* d_in, const int* in_sizes, int n_in,
                              void* d_out, int out_size, void* d_ws, size_t ws_size,
                              hipStream_t stream) {
  const float* x     = (const float*)d_in[0];  // [2,2048,1024]
  const float* A_log = (const float*)d_in[1];  // [1024,16]
  const float* Wdbc  = (const float*)d_in[2];  // [96,1024]
  const float* bdbc  = (const float*)d_in[3];  // [96]
  const float* Wup   = (const float*)d_in[4];  // [1024,64]
  const float* bup   = (const float*)d_in[5];  // [1024]
  float* y = (float*)d_out;                    // [2,2048,1024]

  float* dbc   = (float*)d_ws;                         // [4096,96]
  float* delta = dbc + (size_t)BL_ * DBC_;             // [4096,1024]
  float* aprod = delta + (size_t)BL_ * D_;             // [B*D, CH, N]
  float* hend  = aprod + (size_t)B_ * D_ * CH_ * N_;   // [B*D, CH, N]
  float* hinit = hend  + (size_t)B_ * D_ * CH_ * N_;   // [B*D, CH, N]

  k_dbc<<<64, 128, 0, stream>>>(x, Wdbc, bdbc, dbc);
  k_delta<<<dim3(64, 16), 128, 0, stream>>>(dbc, Wup, bup, delta);

  // chunked scan: 65536 threads for passes 1/3, 32768 for pass 2
  k_pass1<<<256, 256, 0, stream>>>(x, A_log, dbc, delta, aprod, hend);
  k_pass2<<<128, 256, 0, stream>>>(aprod, hend, hinit);
  k_pass3<<<256, 256, 0, stream>>>(x, A_log, dbc, delta, hinit, y);
}